// CutCrossEntropyLoss_73186242724043
// MI455X (gfx1250) — compile-verified
//
#include <hip/hip_runtime.h>
#include <math.h>

// CutCrossEntropyLoss (sampled softmax / NCE-style) for MI455X (gfx1250).
//
// Shapes: hidden [1024, 768] f32, weight [50257, 768] f32,
//         target [1024] i32, noise_indx [1024, 512] i32. Output: scalar f32.
//
// Roofline: ~0.8 GFLOP vs ~1.6 GB of gathered weight rows; weight (154 MB)
// is L2-resident (192 MB L2) -> L2-gather-bandwidth bound. Dot products run
// on the fp32 WMMA pipe (V_WMMA_F32_16X16X4_F32) keeping fp32 numerics.
// Each wave drives TWO independent 16-row tiles -> two independent WMMA
// accumulate chains and 2x the gather loads in flight, sharing one B (h)
// fragment stream from LDS.

typedef __attribute__((ext_vector_type(2))) float v2f;
typedef __attribute__((ext_vector_type(8))) float v8f;

#define HDIM     768
#define SAMPLES  512
#define NROWS    513          // 1 target + 512 noise logits per token
#define NTILES   34           // rows padded to 544 so tiles pair evenly
#define SMOOTH   0.1f

__global__ __launch_bounds__(256) void nce_loss_token_kernel(
    const float* __restrict__ hidden,   // [N, 768]
    const float* __restrict__ weight,   // [50257, 768]
    const int*   __restrict__ target,   // [N]
    const int*   __restrict__ noise,    // [N, 512]
    float*       __restrict__ loss_ws)  // [N] per-token loss
{
    __shared__ float h_sm[HDIM];            // token hidden vector
    __shared__ float logit_sm[NTILES * 16]; // 544 logits (rows >=513 padded)
    __shared__ float red_sm[256];

    const int n    = blockIdx.x;
    const int tid  = threadIdx.x;
    const int lane = tid & 31;
    const int wave = tid >> 5;

    // Stage h[n] into LDS (3 KB), cooperatively.
    for (int i = tid; i < HDIM; i += 256) h_sm[i] = hidden[(size_t)n * HDIM + i];
    __syncthreads();

    // f32 WMMA fragment mapping (16x16x4):
    //   A 16x4: lanes 0-15 hold rows M=0..15 with K={0,1}; lanes 16-31 K={2,3}.
    //   B 4x16: h chunk broadcast over all 16 columns -> every D column holds
    //   the same 16 dot products; lane m = lane&15 selects the gathered row.
    const int m  = lane & 15;
    const int kk = (lane >> 4) * 2;   // this lane's K offset inside a K=4 step

    // Each wave owns tile pairs (t, t+1); 8 waves cover 16 tiles per pass.
    for (int t = wave * 2; t < NTILES; t += 16) {
        const int r0 = t * 16 + m;          // row 0 = target, 1..512 = noise
        const int r1 = r0 + 16;
        int vidx0 = 0, vidx1 = 0;
        if (r0 == 0)            vidx0 = target[n];
        else if (r0 <= SAMPLES) vidx0 = noise[(size_t)n * SAMPLES + (r0 - 1)];
        if (r1 <= SAMPLES)      vidx1 = noise[(size_t)n * SAMPLES + (r1 - 1)];
        const float* rowp0 = weight + (size_t)vidx0 * HDIM + kk;
        const float* rowp1 = weight + (size_t)vidx1 * HDIM + kk;

        v8f c0 = {0.f, 0.f, 0.f, 0.f, 0.f, 0.f, 0.f, 0.f};
        v8f c1 = {0.f, 0.f, 0.f, 0.f, 0.f, 0.f, 0.f, 0.f};
        #pragma unroll 4
        for (int k0 = 0; k0 < HDIM; k0 += 4) {
            v2f a0 = *(const v2f*)(rowp0 + k0);      // gathered W rows (tile t)
            v2f a1 = *(const v2f*)(rowp1 + k0);      // gathered W rows (tile t+1)
            v2f b; b.x = h_sm[k0 + kk]; b.y = h_sm[k0 + kk + 1];
            c0 = __builtin_amdgcn_wmma_f32_16x16x4_f32(
                     false, a0, false, b, (short)0, c0, false, false);
            c1 = __builtin_amdgcn_wmma_f32_16x16x4_f32(
                     false, a1, false, b, (short)0, c1, false, false);
        }

        // Column 0 of D: lane 0 holds M=0..7 in c[0..7], lane 16 holds M=8..15.
        if ((lane & 15) == 0) {
            const int mbase = t * 16 + (lane >> 4) * 8;
            #pragma unroll
            for (int v = 0; v < 8; ++v) {
                logit_sm[mbase + v]      = c0[v];
                logit_sm[mbase + 16 + v] = c1[v];
            }
        }
    }
    __syncthreads();

    // ---- log-softmax over 513 logits + smoothed NLL ----
    float lm = -INFINITY;
    for (int i = tid; i < NROWS; i += 256) lm = fmaxf(lm, logit_sm[i]);
    red_sm[tid] = lm; __syncthreads();
    for (int s = 128; s > 0; s >>= 1) {
        if (tid < s) red_sm[tid] = fmaxf(red_sm[tid], red_sm[tid + s]);
        __syncthreads();
    }
    const float mx = red_sm[0];
    __syncthreads();

    float se = 0.f, ns = 0.f;
    for (int i = tid; i < NROWS; i += 256) {
        const float l = logit_sm[i];
        se += expf(l - mx);
        if (i > 0) ns += l;           // sum of noise logits
    }
    red_sm[tid] = se; __syncthreads();
    for (int s = 128; s > 0; s >>= 1) {
        if (tid < s) red_sm[tid] += red_sm[tid + s];
        __syncthreads();
    }
    const float sumexp = red_sm[0];
    __syncthreads();

    red_sm[tid] = ns; __syncthreads();
    for (int s = 128; s > 0; s >>= 1) {
        if (tid < s) red_sm[tid] += red_sm[tid + s];
        __syncthreads();
    }

    if (tid == 0) {
        const float noise_sum = red_sm[0];
        const float lse  = mx + logf(sumexp);
        // loss = -0.9*(l0-lse) - (0.1/512)*sum(l_s - lse)
        //      = lse - 0.9*l0 - (0.1/512)*sum(l_noise)
        const float loss = lse - (1.0f - SMOOTH) * logit_sm[0]
                               - (SMOOTH / (float)SAMPLES) * noise_sum;
        loss_ws[n] = loss;
    }
}

// Deterministic single-block mean over the N per-token losses.
__global__ __launch_bounds__(256) void nce_loss_mean_kernel(
    const float* __restrict__ loss_ws, float* __restrict__ out, int N)
{
    __shared__ float red_sm[256];
    const int tid = threadIdx.x;
    float s = 0.f;
    for (int i = tid; i < N; i += 256) s += loss_ws[i];
    red_sm[tid] = s; __syncthreads();
    for (int k = 128; k > 0; k >>= 1) {
        if (tid < k) red_sm[tid] += red_sm[tid + k];
        __syncthreads();
    }
    if (tid == 0) out[0] = red_sm[0] / (float)N;
}

extern "C" void kernel_launch(void* const* d_in, const int* in_sizes, int n_in,
                              void* d_out, int out_size, void* d_ws, size_t ws_size,
                              hipStream_t stream) {
    const float* hidden = (const float*)d_in[0];   // [2,512,768] f32
    const float* weight = (const float*)d_in[1];   // [50257,768] f32
    const int*   target = (const int*)d_in[2];     // [2,512] int
    const int*   noise  = (const int*)d_in[3];     // [1024,512] int
    float* out = (float*)d_out;                    // scalar f32
    float* ws  = (float*)d_ws;                     // >= 1024 floats

    const int N = in_sizes[2];                     // 1024 tokens

    nce_loss_token_kernel<<<N, 256, 0, stream>>>(hidden, weight, target, noise, ws);
    nce_loss_mean_kernel<<<1, 256, 0, stream>>>(ws, out, N);
}